// EGNNLayer_2319282340045
// MI455X (gfx1250) — compile-verified
//
#include <hip/hip_runtime.h>
#include <math.h>

// ---------------- problem constants (match reference) ----------------
#define N_NODES   32768
#define N_EDGES   524288
#define NODE_DIM  128
#define HIDDEN_DIM 256
#define EDGE_DIM  64
#define IN_DIM    320          // 2*NODE_DIM + EDGE_DIM

#define TILE_E    64           // edges per block
#define THREADS   256          // 8 wave32

// ---------------- LDS layout (float offsets) ----------------
#define A_LD      324          // 320 + pad4  -> stride%64==4, conflict-free b64 frags
#define A_OFF     0
#define A_SIZE    (TILE_E * A_LD)          // 20736
#define H_LD      260          // 256 + pad4
#define H_OFF     (A_OFF + A_SIZE)
#define H_SIZE    (TILE_E * H_LD)          // 16640
#define B_OFF     (H_OFF + H_SIZE)         // K-panel staging, up to 32x256 packed
#define B_SIZE    (32 * 256)               // 8192
#define S_OFF     (B_OFF + B_SIZE)
#define SMEM_FLOATS (S_OFF + 4 * TILE_E)   // src/dst/dist + spare
// total = 45824 floats = 183296 bytes  (< 320KB CDNA5 WGP LDS)

typedef float v2f __attribute__((ext_vector_type(2)));
typedef float v8f __attribute__((ext_vector_type(8)));

__device__ __forceinline__ float silu_f(float v) {
    return v / (1.0f + __expf(-v));
}

__device__ __forceinline__ void atomAddF(float* p, float v) {
    __hip_atomic_fetch_add(p, v, __ATOMIC_RELAXED, __HIP_MEMORY_SCOPE_AGENT);
}

// ------------------------------------------------------------------
// Block-cooperative GEMM over a 64-row LDS A tile.
// Each wave computes a 16(M) x (NTILES*16)(N) slab; wave pairs split the
// staged N range (nsel = wave&1 picks columns [nsel*NSTAGE/2 ...)).
// Weights are staged per 32-deep K panel into Bs, pre-packed so a lane's
// B fragment is one aligned 8-byte LDS load:
//   Bs[(k/4)*NSTAGE*4 + n*4 + (k&3)] = W[k][n]
//   lane<16 reads k%4 = {0,1}, lane>=16 reads {2,3}   (per ISA B layout)
// A fragment per ISA: lanes0-15 K={0,1}, lanes16-31 K={2,3} of row (lane&15).
// ------------------------------------------------------------------
template<int NTILES, int NSTAGE>
__device__ __forceinline__ void gemm_tile(const float* __restrict__ W, int ldw, int K,
                                          const float* __restrict__ A, int lda,
                                          float* __restrict__ Bs,
                                          v8f* acc, int tid)
{
    const int lane  = tid & 31;
    const int wave  = tid >> 5;
    const int row   = lane & 15;
    const int hi    = lane >> 4;         // 0: K={0,1}, 1: K={2,3}
    const int etile = wave >> 1;         // which 16-edge M tile
    const int nsel  = wave & 1;          // which half of staged N
    const int nbase = nsel * (NSTAGE / 2);

    for (int kp = 0; kp < K; kp += 32) {
        __syncthreads();                                   // Bs reuse fence
        for (int i = tid; i < 32 * NSTAGE; i += THREADS) { // stage + pack panel
            int kl = i / NSTAGE;
            int n  = i - kl * NSTAGE;
            Bs[(((kl >> 2) * NSTAGE + n) << 2) + (kl & 3)] = W[(kp + kl) * ldw + n];
        }
        __syncthreads();

        const float* Arow = A + (etile * 16 + row) * lda + kp + 2 * hi;
#pragma unroll
        for (int kl = 0; kl < 32; kl += 4) {
            v2f a = *(const v2f*)(Arow + kl);              // ds_load_b64
#pragma unroll
            for (int nt = 0; nt < NTILES; ++nt) {
                v2f b = *(const v2f*)(Bs + ((((kl >> 2) * NSTAGE + nbase + nt * 16 + row) << 2) + 2 * hi));
                acc[nt] = __builtin_amdgcn_wmma_f32_16x16x4_f32(
                    /*neg_a=*/false, a, /*neg_b=*/false, b,
                    /*c_mod=*/(short)0, acc[nt],
                    /*reuse_a=*/false, /*reuse_b=*/false);
            }
        }
    }
}

// ------------------------------------------------------------------
// out[0 : N*128]        = h   (residual base)
// out[N*128 : N*128+N*3] = x
// ------------------------------------------------------------------
__global__ void egnn_init_out(const float* __restrict__ h,
                              const float* __restrict__ x,
                              float* __restrict__ out)
{
    int i = blockIdx.x * blockDim.x + threadIdx.x;
    if (i < N_NODES * NODE_DIM)                out[i] = h[i];
    else if (i < N_NODES * (NODE_DIM + 3))     out[i] = x[i - N_NODES * NODE_DIM];
}

// ------------------------------------------------------------------
// Main fused edge kernel: 64 edges / block.
// ------------------------------------------------------------------
__global__ void __launch_bounds__(THREADS)
egnn_edge_kernel(const float* __restrict__ h,   const float* __restrict__ x,
                 const int*   __restrict__ eidx,const float* __restrict__ edist,
                 const float* __restrict__ nW1, const float* __restrict__ nb1,
                 const float* __restrict__ nW2, const float* __restrict__ nb2,
                 const float* __restrict__ cW1, const float* __restrict__ cb1,
                 const float* __restrict__ cW2,
                 const float* __restrict__ eW1, const float* __restrict__ eb1,
                 const float* __restrict__ eW2, const float* __restrict__ eb2,
                 float* __restrict__ out)
{
    extern __shared__ float smem[];
    float* Ablk = smem + A_OFF;              // [64][324] m_input tile
    float* Hblk = smem + H_OFF;              // [64][260] silu(hidden) / edge-mlp scratch
    float* Bs   = smem + B_OFF;              // packed weight panel
    int*   SRCs = (int*)(smem + S_OFF);
    int*   DSTs = (int*)(smem + S_OFF + TILE_E);
    float* DIST = smem + S_OFF + 2 * TILE_E;

    const int tid   = threadIdx.x;
    const int lane  = tid & 31;
    const int wave  = tid >> 5;
    const int row   = lane & 15;
    const int hi    = lane >> 4;
    const int etile = wave >> 1;
    const int nsel  = wave & 1;
    const int e0    = blockIdx.x * TILE_E;

    if (tid < TILE_E) {
        SRCs[tid] = eidx[e0 + tid];
        DSTs[tid] = eidx[N_EDGES + e0 + tid];
        DIST[tid] = edist[e0 + tid];
    }
    __syncthreads();

    // ---- gather h[src] -> A[:,0:128], h[dst] -> A[:,128:256]  (float4) ----
    for (int i = tid; i < TILE_E * 64; i += THREADS) {
        int e = i >> 6, q = i & 63;
        int half = q >> 5, c4 = q & 31;
        int node = half ? DSTs[e] : SRCs[e];
        *(float4*)(Ablk + e * A_LD + half * NODE_DIM + c4 * 4) =
            ((const float4*)h)[node * (NODE_DIM / 4) + c4];
    }

    // ---- edge MLP: hid[k][e] = silu(dist*eW1 + eb1)  (uses Hblk as scratch) ----
    float* hid = Hblk;
    for (int i = tid; i < TILE_E * EDGE_DIM; i += THREADS) {
        int e = i >> 6, k = i & 63;
        hid[k * TILE_E + e] = silu_f(DIST[e] * eW1[k] + eb1[k]);
    }
    __syncthreads();
    // attr = hid^T @ eW2 + eb2  -> A[:,256:320]
    for (int i = tid; i < TILE_E * EDGE_DIM; i += THREADS) {
        int e = i >> 6, j = i & 63;
        float s = eb2[j];
#pragma unroll 8
        for (int k = 0; k < EDGE_DIM; ++k)
            s += hid[k * TILE_E + e] * eW2[k * EDGE_DIM + j];  // hid read is wave-uniform
        Ablk[e * A_LD + 2 * NODE_DIM + j] = s;
    }
    // (A-tile complete; gemm_tile's first barrier orders all readers)

    // ================= GEMM1: H = silu(A @ nW1 + nb1) =================
    {
        v8f acc[8];
#pragma unroll
        for (int nt = 0; nt < 8; ++nt) {
            float bv = nb1[nsel * 128 + nt * 16 + row];
            v8f t = {bv, bv, bv, bv, bv, bv, bv, bv};
            acc[nt] = t;
        }
        gemm_tile<8, 256>(nW1, HIDDEN_DIM, IN_DIM, Ablk, A_LD, Bs, acc, tid);
#pragma unroll
        for (int nt = 0; nt < 8; ++nt)
#pragma unroll
            for (int i = 0; i < 8; ++i) {              // C layout: M = i + 8*hi, N = row
                int e = etile * 16 + hi * 8 + i;
                Hblk[e * H_LD + nsel * 128 + nt * 16 + row] = silu_f(acc[nt][i]);
            }
    }

    // ================= GEMM2: m = H @ nW2 + nb2 ; scatter-add =================
    {
        v8f acc[4];
#pragma unroll
        for (int nt = 0; nt < 4; ++nt) {
            float bv = nb2[nsel * 64 + nt * 16 + row];
            v8f t = {bv, bv, bv, bv, bv, bv, bv, bv};
            acc[nt] = t;
        }
        gemm_tile<4, 128>(nW2, NODE_DIM, HIDDEN_DIM, Hblk, H_LD, Bs, acc, tid);
#pragma unroll
        for (int nt = 0; nt < 4; ++nt)
#pragma unroll
            for (int i = 0; i < 8; ++i) {
                int e = etile * 16 + hi * 8 + i;
                int d = DSTs[e];
                atomAddF(out + d * NODE_DIM + nsel * 64 + nt * 16 + row, acc[nt][i]);
            }
    }

    // ================= GEMM3: G = silu(A @ cW1 + cb1)  (reuses Hblk) =================
    {
        v8f acc[8];
#pragma unroll
        for (int nt = 0; nt < 8; ++nt) {
            float bv = cb1[nsel * 128 + nt * 16 + row];
            v8f t = {bv, bv, bv, bv, bv, bv, bv, bv};
            acc[nt] = t;
        }
        gemm_tile<8, 256>(cW1, HIDDEN_DIM, IN_DIM, Ablk, A_LD, Bs, acc, tid);
#pragma unroll
        for (int nt = 0; nt < 8; ++nt)
#pragma unroll
            for (int i = 0; i < 8; ++i) {
                int e = etile * 16 + hi * 8 + i;
                Hblk[e * H_LD + nsel * 128 + nt * 16 + row] = silu_f(acc[nt][i]);
            }
    }
    __syncthreads();

    // ================= coordinate update =================
    if (tid < TILE_E) {
        int e = tid;
        float s = 0.0f;
#pragma unroll 8
        for (int k = 0; k < HIDDEN_DIM; ++k)
            s += Hblk[e * H_LD + k] * cW2[k];          // coord_weight (no bias)
        int si = SRCs[e], di = DSTs[e];
        float dx = x[si * 3 + 0] - x[di * 3 + 0];
        float dy = x[si * 3 + 1] - x[di * 3 + 1];
        float dz = x[si * 3 + 2] - x[di * 3 + 2];
        float len = fmaxf(sqrtf(dx * dx + dy * dy + dz * dz), 1e-8f);
        float w = s / len;
        float* ox = out + N_NODES * NODE_DIM;
        atomAddF(ox + di * 3 + 0, w * dx);
        atomAddF(ox + di * 3 + 1, w * dy);
        atomAddF(ox + di * 3 + 2, w * dz);
    }
}

// ------------------------------------------------------------------
extern "C" void kernel_launch(void* const* d_in, const int* in_sizes, int n_in,
                              void* d_out, int out_size, void* d_ws, size_t ws_size,
                              hipStream_t stream)
{
    const float* h     = (const float*)d_in[0];
    const float* x     = (const float*)d_in[1];
    const int*   eidx  = (const int*)  d_in[2];
    const float* edist = (const float*)d_in[3];
    const float* nW1   = (const float*)d_in[4];
    const float* nb1   = (const float*)d_in[5];
    const float* nW2   = (const float*)d_in[6];
    const float* nb2   = (const float*)d_in[7];
    const float* cW1   = (const float*)d_in[8];
    const float* cb1   = (const float*)d_in[9];
    const float* cW2   = (const float*)d_in[10];
    const float* eW1   = (const float*)d_in[11];
    const float* eb1   = (const float*)d_in[12];
    const float* eW2   = (const float*)d_in[13];
    const float* eb2   = (const float*)d_in[14];
    float* out = (float*)d_out;

    const int tot = N_NODES * (NODE_DIM + 3);
    egnn_init_out<<<(tot + 255) / 256, 256, 0, stream>>>(h, x, out);

    const size_t smem_bytes = (size_t)SMEM_FLOATS * sizeof(float);
    egnn_edge_kernel<<<N_EDGES / TILE_E, THREADS, smem_bytes, stream>>>(
        h, x, eidx, edist,
        nW1, nb1, nW2, nb2,
        cW1, cb1, cW2,
        eW1, eb1, eW2, eb2,
        out);
}